// QuantumEmbedding_13554916786258
// MI455X (gfx1250) — compile-verified
//
#include <hip/hip_runtime.h>
#include <hip/hip_bf16.h>

typedef float v2f __attribute__((ext_vector_type(2)));
typedef float v8f __attribute__((ext_vector_type(8)));

#define TPW 4                 // 16-sample tiles per wave
#define WAVES_PER_BLOCK 8
#define THREADS (WAVES_PER_BLOCK * 32)

__device__ __forceinline__ float fast_tanh(float x) {
#if __has_builtin(__builtin_amdgcn_tanhf)
    return __builtin_amdgcn_tanhf(x);
#else
    return tanhf(x);
#endif
}

// xor-distance permute within each 16-lane row, as pure VALU (v_permlane16_b32).
// Selector nibble for lane i is (i ^ D), packed 8 lanes per 32-bit word.
template <int D, unsigned S0, unsigned S1>
__device__ __forceinline__ float lane_xor16(float v) {
#if __has_builtin(__builtin_amdgcn_permlane16)
    unsigned u = __float_as_uint(v);
    u = __builtin_amdgcn_permlane16(u, u, S0, S1, false, false);
    return __uint_as_float(u);
#else
    return __shfl_xor(v, D, 32);
#endif
}

// ---- tiny 4-qubit gate helpers (qubit q <-> bit (3-q) of flat index) ----
__device__ __forceinline__ void apply_ry(float* sr, float* si, int q, float th) {
    float s, c;
    __sincosf(th * 0.5f, &s, &c);
    const int mask = 1 << (3 - q);
#pragma unroll
    for (int k = 0; k < 16; ++k) {
        if (!(k & mask)) {
            const int k1 = k | mask;
            float ar = sr[k], ai = si[k], br = sr[k1], bi = si[k1];
            sr[k]  = c * ar - s * br;  si[k]  = c * ai - s * bi;
            sr[k1] = s * ar + c * br;  si[k1] = s * ai + c * bi;
        }
    }
}

__device__ __forceinline__ void apply_rz(float* sr, float* si, int q, float th) {
    float s, c;
    __sincosf(th * 0.5f, &s, &c);
    const int mask = 1 << (3 - q);
#pragma unroll
    for (int k = 0; k < 16; ++k) {
        float re = sr[k], im = si[k];
        if (k & mask) {           // * exp(+i th/2)
            sr[k] = re * c - im * s;  si[k] = im * c + re * s;
        } else {                  // * exp(-i th/2)
            sr[k] = re * c + im * s;  si[k] = im * c - re * s;
        }
    }
}

__device__ __forceinline__ void apply_cnot(float* sr, float* si, int c, int t) {
    const int mc = 1 << (3 - c), mt = 1 << (3 - t);
#pragma unroll
    for (int k = 0; k < 16; ++k) {
        if ((k & mc) && !(k & mt)) {
            const int k1 = k | mt;
            float tr = sr[k], ti = si[k];
            sr[k] = sr[k1];  si[k] = si[k1];
            sr[k1] = tr;     si[k1] = ti;
        }
    }
}

__global__ __launch_bounds__(THREADS)
void qc4_wmma_kernel(const float* __restrict__ x,
                     const float* __restrict__ w,
                     float* __restrict__ out) {
    // B-matrix (U^T) staged through LDS: row j (=K dim) holds column j of U.
    __shared__ float sBre[256];
    __shared__ float sBim[256];

    const int tid = threadIdx.x;

    // ---- build U (16 columns) once per block; thread j simulates basis |j> ----
    if (tid < 16) {
        float sr[16], si[16];
#pragma unroll
        for (int k = 0; k < 16; ++k) { sr[k] = (k == tid) ? 1.0f : 0.0f; si[k] = 0.0f; }
#pragma unroll
        for (int layer = 0; layer < 2; ++layer) {
#pragma unroll
            for (int q = 0; q < 4; ++q) {
                apply_ry(sr, si, q, w[(layer * 4 + q) * 2 + 0]);
                apply_rz(sr, si, q, w[(layer * 4 + q) * 2 + 1]);
            }
            apply_cnot(sr, si, 0, 1);
            apply_cnot(sr, si, 1, 2);
            apply_cnot(sr, si, 2, 3);
            apply_cnot(sr, si, 3, 0);
        }
#pragma unroll
        for (int k = 0; k < 16; ++k) {
            sBre[tid * 16 + k] = sr[k];   // B[j][k] = U[k][j]
            sBim[tid * 16 + k] = si[k];
        }
    }
    __syncthreads();

    const int lane = tid & 31;
    const int h    = lane >> 4;   // wave half (selects K pair within 4-K slice)
    const int n    = lane & 15;   // A row / B,D column within half

    // ---- load B fragments (4 K-slices x {re,im}) once per wave ----
    // 32-bit 4x16 B layout: lanes 0-15 -> K = {0,1}, lanes 16-31 -> K = {2,3}; N = lane%16
    v2f bre[4], bim[4];
#pragma unroll
    for (int kb = 0; kb < 4; ++kb) {
        const int k0 = 4 * kb + 2 * h;
        v2f r = { sBre[k0 * 16 + n], sBre[(k0 + 1) * 16 + n] };
        v2f i = { sBim[k0 * 16 + n], sBim[(k0 + 1) * 16 + n] };
        bre[kb] = r;
        bim[kb] = i;
    }

    // per-lane WHT butterfly signs: sg[i] = -1 if (n & (1<<i)) else +1
    float sg[4];
#pragma unroll
    for (int i = 0; i < 4; ++i) sg[i] = (n & (1 << i)) ? -1.0f : 1.0f;

    // output role of this lane (hoisted out of the tile loop):
    // WHT index 1<<(3-q) holds z_q -> lanes n in {8,4,2,1} write q in {0,1,2,3}
    const bool writer = (n == 1) || (n == 2) || (n == 4) || (n == 8);
    const int  q      = (n == 8) ? 0 : (n == 4) ? 1 : (n == 2) ? 2 : 3;

    const int wave = (blockIdx.x * (int)blockDim.x + tid) >> 5;

#pragma unroll 1
    for (int t = 0; t < TPW; ++t) {
        const int tile = wave * TPW + t;
        const int row_sample = tile * 16 + n;      // this lane's A-matrix row

        // ---- embedding amplitudes per qubit: (cos,sin)(tanh(x)*pi/2) ----
        const float4 xv = *(const float4*)(x + (size_t)row_sample * 8);
        float cq[4], sq[4];
        {
            float xa[4] = { xv.x, xv.y, xv.z, xv.w };
#pragma unroll
            for (int qq = 0; qq < 4; ++qq) {
                float a = fast_tanh(xa[qq]) * 1.5707963267948966f;
                __sincosf(a, &sq[qq], &cq[qq]);
            }
        }
        // E[4kb+2h+j] = m01[kb] * (h? s2:c2) * (j? s3:c3)  -- no dynamic indexing
        float m01[4] = { cq[0] * cq[1], cq[0] * sq[1], sq[0] * cq[1], sq[0] * sq[1] };
        const float u2 = h ? sq[2] : cq[2];
        const float t0 = u2 * cq[3];
        const float t1 = u2 * sq[3];

        // ---- psi = E x U^T : chained fp32 WMMA, K = 16 as 4 slices of 4 ----
        v8f accre = { 0, 0, 0, 0, 0, 0, 0, 0 };
        v8f accim = { 0, 0, 0, 0, 0, 0, 0, 0 };
#pragma unroll
        for (int kb = 0; kb < 4; ++kb) {
            v2f a = { m01[kb] * t0, m01[kb] * t1 };
            accre = __builtin_amdgcn_wmma_f32_16x16x4_f32(
                false, a, false, bre[kb], (short)0, accre, false, false);
            accim = __builtin_amdgcn_wmma_f32_16x16x4_f32(
                false, a, false, bim[kb], (short)0, accim, false, false);
        }

        // ---- p = |psi|^2 ; z_q = WHT_16(p) at index 1<<(3-q) ----
        // C/D layout: lane(h,n), VGPR v -> sample (v + 8h), state index n.
        float z[8];
#pragma unroll
        for (int v = 0; v < 8; ++v) {
            float p = accre[v] * accre[v] + accim[v] * accim[v];
            p = fmaf(sg[0], p, lane_xor16<1, 0x67452301u, 0xEFCDAB89u>(p));
            p = fmaf(sg[1], p, lane_xor16<2, 0x54761032u, 0xDCFE98BAu>(p));
            p = fmaf(sg[2], p, lane_xor16<4, 0x32107654u, 0xBA98FEDCu>(p));
            p = fmaf(sg[3], p, lane_xor16<8, 0xFEDCBA98u, 0x76543210u>(p));
            z[v] = p;
        }
        if (writer) {
            float* dst = out + (size_t)tile * 64 + (32 * h + q);
#pragma unroll
            for (int v = 0; v < 8; ++v) dst[4 * v] = z[v];
        }
    }
}

extern "C" void kernel_launch(void* const* d_in, const int* in_sizes, int n_in,
                              void* d_out, int out_size, void* d_ws, size_t ws_size,
                              hipStream_t stream) {
    (void)n_in; (void)out_size; (void)d_ws; (void)ws_size;
    const float* x = (const float*)d_in[0];     // [B, 8] fp32
    const float* w = (const float*)d_in[1];     // [2, 4, 2] fp32
    float* out = (float*)d_out;                 // [B, 4] fp32

    const int B = in_sizes[0] / 8;              // 524288
    const int tiles = B / 16;                   // 32768 (B = 2^19, exact)
    const int waves = tiles / TPW;              // 8192
    const int blocks = (waves * 32) / THREADS;  // 1024

    qc4_wmma_kernel<<<blocks, THREADS, 0, stream>>>(x, w, out);
}